// SparseLoss_62294205661187
// MI455X (gfx1250) — compile-verified
//
#include <hip/hip_runtime.h>
#include <hip/hip_bf16.h>

typedef __attribute__((ext_vector_type(2))) float v2f;
typedef __attribute__((ext_vector_type(8))) float v8f;

#define WAVES   4
#define KC      32                 // K-chunk staged in LDS per iteration
#define STR     36                 // KC + 4 pad -> bank-conflict-free strided reads
#define DDIM    1024
#define NROWS   16384
#define NTILES  (NROWS / 16)
#define NCHUNK  (DDIM / KC)

// Low 32 bits of a flat shared-memory pointer == workgroup-relative LDS byte
// address (flat LDS addresses are {shared-aperture, lds_offset}).
__device__ __forceinline__ unsigned lds_lo32(const void* p) {
    return (unsigned)(size_t)p;
}

// CDNA5 async global->LDS copy (ASYNCcnt-tracked, no VGPR round-trip).
__device__ __forceinline__ void async_b128(unsigned ldsb, const float* g) {
    asm volatile("global_load_async_to_lds_b128 %0, %1, off"
                 :: "v"(ldsb), "v"((unsigned long long)(size_t)g)
                 : "memory");
}
// Non-temporal variant for the streamed-once x matrix (keep L2 for recovers).
__device__ __forceinline__ void async_b128_nt(unsigned ldsb, const float* g) {
    asm volatile("global_load_async_to_lds_b128 %0, %1, off th:TH_LOAD_NT"
                 :: "v"(ldsb), "v"((unsigned long long)(size_t)g)
                 : "memory");
}
__device__ __forceinline__ void wait_async_8() {   // older chunk (8 ops) done
    asm volatile("s_wait_asynccnt 0x8" ::: "memory");
}
__device__ __forceinline__ void wait_async_0() {   // everything done
    asm volatile("s_wait_asynccnt 0x0" ::: "memory");
}

// One wave32 per 16-row tile. Double-buffered async staging of 16xKC tiles of
// x and gathered y into LDS, then KC/4 f32 WMMAs per chunk with B == ones to
// reduce the K dimension into the f32 accumulator (D[m,n] += sum_k A[m,k]).
__global__ __launch_bounds__(WAVES * 32) void
sparse_loss_tile_kernel(const float* __restrict__ x,
                        const float* __restrict__ rec,
                        const long long* __restrict__ lbl,
                        float* __restrict__ tileout) {
    __shared__ float xs[2][WAVES][16 * STR];
    __shared__ float ys[2][WAVES][16 * STR];

    const int tid  = threadIdx.x;
    const int wave = tid >> 5;
    const int lane = tid & 31;
    const int tile = blockIdx.x * WAVES + wave;
    const int row0 = tile << 4;

    const int h   = lane >> 4;       // half-wave: 0 or 1
    const int l16 = lane & 15;

    // Staging map: per iteration 32 lanes cover 4 rows x 8 lanes x 4 floats.
    const int sub  = lane & 7;       // lane within row
    const int rg   = lane >> 3;      // row group 0..3
    const int scol = sub << 2;       // 0..28

    // Row pointers (x row + gathered recovers row) for the 4 rows this thread
    // stages. Labels are int64 in the reference.
    const float* xrp[4];
    const float* yrp[4];
#pragma unroll
    for (int it = 0; it < 4; ++it) {
        const int r = (it << 2) + rg;
        xrp[it] = x + (size_t)(row0 + r) * DDIM + scol;
        yrp[it] = rec + (size_t)lbl[row0 + r] * DDIM + scol;
    }

    unsigned xl[2], yl[2];
#pragma unroll
    for (int b = 0; b < 2; ++b) {
        xl[b] = lds_lo32(&xs[b][wave][0]);
        yl[b] = lds_lo32(&ys[b][wave][0]);
    }
    const unsigned sofs = (unsigned)(scol * 4);

    // Prologue: issue chunk 0 into buffer 0 (8 async b128 ops per thread).
#pragma unroll
    for (int it = 0; it < 4; ++it) {
        const unsigned ro = (unsigned)(((it << 2) + rg) * STR * 4) + sofs;
        async_b128_nt(xl[0] + ro, xrp[it]);
        async_b128  (yl[0] + ro, yrp[it]);
    }

    v8f acc = {};                                   // 16x16 f32 row-sum acc
    v2f bones; bones[0] = 1.0f; bones[1] = 1.0f;    // B = ones(4x16)

    for (int i = 0; i < NCHUNK; ++i) {
        if (i + 1 < NCHUNK) {
            // Issue next chunk into the other buffer, then wait for chunk i.
            const int kc = (i + 1) * KC;
            const int b  = (i + 1) & 1;
#pragma unroll
            for (int it = 0; it < 4; ++it) {
                const unsigned ro = (unsigned)(((it << 2) + rg) * STR * 4) + sofs;
                async_b128_nt(xl[b] + ro, xrp[it] + kc);
                async_b128  (yl[b] + ro, yrp[it] + kc);
            }
            wait_async_8();
        } else {
            wait_async_0();
        }
        __syncthreads();

        const float* xw = &xs[i & 1][wave][0];
        const float* yw = &ys[i & 1][wave][0];
        // A layout (16x4 f32): VGPR0 = {lanes 0-15: (M=lane,K=0), 16-31: K=2}
        //                      VGPR1 = {K=1, K=3}
#pragma unroll
        for (int ks = 0; ks < KC; ks += 4) {
            const int c0 = ks + (h << 1);
            const float2 xv = *(const float2*)&xw[l16 * STR + c0];
            const float2 yv = *(const float2*)&yw[l16 * STR + c0];
            const float d0 = xv.x - yv.x;
            const float d1 = xv.y - yv.y;
            v2f a; a[0] = d0 * d0; a[1] = d1 * d1;
            acc = __builtin_amdgcn_wmma_f32_16x16x4_f32(
                false, a, false, bones, (short)0, acc, false, false);
        }
        __syncthreads();
    }

    // acc[j]: lanes 0-15 hold rowsum(M=j), lanes 16-31 hold rowsum(M=8+j),
    // identical across columns (B was all ones).
    float s = 0.0f;
#pragma unroll
    for (int j = 0; j < 8; ++j) s += sqrtf(acc[j]);

    const float t0  = __shfl(s, 0, 32);    // rows 0..7
    const float t16 = __shfl(s, 16, 32);   // rows 8..15
    if (lane == 0) tileout[tile] = t0 + t16;
}

// Deterministic final reduction of the 1024 per-tile partial sums.
__global__ __launch_bounds__(256) void
sparse_loss_reduce_kernel(const float* __restrict__ tileout,
                          float* __restrict__ out) {
    __shared__ float sm[256];
    float s = 0.0f;
    for (int i = threadIdx.x; i < NTILES; i += 256) s += tileout[i];
    sm[threadIdx.x] = s;
    __syncthreads();
    if (threadIdx.x == 0) {
        float t = 0.0f;
        for (int i = 0; i < 256; ++i) t += sm[i];
        out[0] = t / (float)NROWS;
    }
}

extern "C" void kernel_launch(void* const* d_in, const int* in_sizes, int n_in,
                              void* d_out, int out_size, void* d_ws, size_t ws_size,
                              hipStream_t stream) {
    (void)in_sizes; (void)n_in; (void)out_size; (void)ws_size;
    const float*     x   = (const float*)d_in[0];      // [16384, 1024] f32
    const float*     rec = (const float*)d_in[1];      // [ 8192, 1024] f32
    const long long* lbl = (const long long*)d_in[2];  // [16384] int64
    float* out = (float*)d_out;                        // scalar f32
    float* ws  = (float*)d_ws;                         // >= 1024 floats

    sparse_loss_tile_kernel<<<NTILES / WAVES, WAVES * 32, 0, stream>>>(x, rec, lbl, ws);
    sparse_loss_reduce_kernel<<<1, 256, 0, stream>>>(ws, out);
}